// MaskRenderer_24051816858186
// MI455X (gfx1250) — compile-verified
//
#include <hip/hip_runtime.h>

// ---------------------------------------------------------------------------
// MaskRenderer (soft silhouette) for gfx1250 (MI455X, wave32, WMMA).
//
// Per triangle, the 8 per-pixel quantities needed by the edge-distance math
// (u0,u1,u2 = -2 * w.e dots; x0,x1,x2 = edge crosses; |q-a|^2, |q-b|^2) are
// linear in the basis (1, qx, qy, qx^2+qy^2).  V_WMMA_F32_16X16X4_F32 with
// A = [16 quantity rows x 4 coeffs] (2 triangles) and B = [4 basis x 16
// pixels] lands one (pixel, triangle) pair per lane with its 8 quantities in
// d[0..7], so the nonlinear tail is entirely lane-local.
// ---------------------------------------------------------------------------

typedef __attribute__((ext_vector_type(2))) float v2f;
typedef __attribute__((ext_vector_type(4))) float v4f;
typedef __attribute__((ext_vector_type(8))) float v8f;

#define IMG   256
#define NVERT 778
#define NF    1536
#define NB    4
#define PAIRS (NF / 2)
#define COEF_FLOATS  (NF * 32)                      // 8 rows * 4 coeffs per tri
#define CONST_FLOATS (NF * 8)                       // ee0..2, ivn0..2 (+pad)
#define BATCH_FLOATS (COEF_FLOATS + CONST_FLOATS)   // 61440 floats = 240 KB

#define EXP_SCALE  (-14426.950408889634f)           // -(1/SIGMA) * log2(e)
#define OUT_SCALE  (-8.656170245333781f)            // -6 * log2(e)

// ---------------------------------------------------------------------------
// Prep: per (batch, face) build the WMMA A-matrix coefficient rows + per-
// triangle constants.  ws layout per batch (contiguous 61440 floats):
//   [f*32 .. f*32+31]  : 8 rows x (c_const, c_qx, c_qy, c_q2)
//   [COEF_FLOATS + f*8]: ee0, ee1, ee2, ivn0, ivn1, ivn2, 0, 0
// Rows 0..2 carry u_i = -2 * (q-p_i).e_i, and ivn_i = -0.5 / max(ee,1e-12),
// so t = clamp(u*ivn) and d2 = fma(t, fma(t, ee, u), w2) with no rescaling.
// ---------------------------------------------------------------------------
__global__ void prep_kernel(const float* __restrict__ verts,
                            const int* __restrict__ faces,
                            const int* __restrict__ scale_flag,
                            float* __restrict__ ws) {
    int tid = blockIdx.x * blockDim.x + threadIdx.x;
    if (tid >= NB * NF) return;
    int b = tid / NF;
    int f = tid - b * NF;
    const bool sc = scale_flag[0] != 0;

    float vx[3], vy[3];
#pragma unroll
    for (int k = 0; k < 3; ++k) {
        int vi = faces[f * 3 + k];
        float x = verts[(b * NVERT + vi) * 3 + 0];
        float y = verts[(b * NVERT + vi) * 3 + 1];
        if (sc) { x = fmaf(x, 2.0f, -1.0f); y = fmaf(y, 2.0f, -1.0f); }
        vx[k] = -x;   // v2 = (-vx, -vy)
        vy[k] = -y;
    }
    float ax = vx[0], ay = vy[0];
    float bx = vx[1], by = vy[1];
    float cx = vx[2], cy = vy[2];

    float e0x = bx - ax, e0y = by - ay;   // edge a->b
    float e1x = cx - bx, e1y = cy - by;   // edge b->c
    float e2x = ax - cx, e2y = ay - cy;   // edge c->a
    float ee0 = e0x * e0x + e0y * e0y;
    float ee1 = e1x * e1x + e1y * e1y;
    float ee2 = e2x * e2x + e2y * e2y;
    float ivn0 = -0.5f / fmaxf(ee0, 1e-12f);
    float ivn1 = -0.5f / fmaxf(ee1, 1e-12f);
    float ivn2 = -0.5f / fmaxf(ee2, 1e-12f);

    float* cf = ws + (size_t)b * BATCH_FLOATS + f * 32;
    // rows 0..2 : u_i = -2[(q.e_i) - p_i.e_i]
    cf[0]  = 2.0f * (ax * e0x + ay * e0y); cf[1]  = -2.0f * e0x; cf[2]  = -2.0f * e0y; cf[3]  = 0.0f;
    cf[4]  = 2.0f * (bx * e1x + by * e1y); cf[5]  = -2.0f * e1x; cf[6]  = -2.0f * e1y; cf[7]  = 0.0f;
    cf[8]  = 2.0f * (cx * e2x + cy * e2y); cf[9]  = -2.0f * e2x; cf[10] = -2.0f * e2y; cf[11] = 0.0f;
    // rows 3..5 : cross_i = e_i x (q - p_i)
    cf[12] = e0y * ax - e0x * ay;          cf[13] = -e0y;        cf[14] = e0x;         cf[15] = 0.0f;
    cf[16] = e1y * bx - e1x * by;          cf[17] = -e1y;        cf[18] = e1x;         cf[19] = 0.0f;
    cf[20] = e2y * cx - e2x * cy;          cf[21] = -e2y;        cf[22] = e2x;         cf[23] = 0.0f;
    // rows 6..7 : |q-p|^2 = q^2 - 2 q.p + |p|^2 (uses the q^2 basis slot)
    cf[24] = ax * ax + ay * ay;            cf[25] = -2.0f * ax;  cf[26] = -2.0f * ay;  cf[27] = 1.0f;
    cf[28] = bx * bx + by * by;            cf[29] = -2.0f * bx;  cf[30] = -2.0f * by;  cf[31] = 1.0f;

    float* cb = ws + (size_t)b * BATCH_FLOATS + COEF_FLOATS + f * 8;
    cb[0] = ee0;  cb[1] = ee1;  cb[2] = ee2;  cb[3] = ivn0;
    cb[4] = ivn1; cb[5] = ivn2; cb[6] = 0.0f; cb[7] = 0.0f;
}

// ---------------------------------------------------------------------------
// Main kernel: 512 blocks x 1024 threads (32 waves).  blockIdx/128 = batch;
// each wave owns 16 consecutive pixels of one image row and loops over the
// 768 triangle pairs.  Triangle tables live in 240 KB of LDS.  Waves whose
// 32 running products are all exactly zero exit early (result invariant).
// ---------------------------------------------------------------------------
__global__ __launch_bounds__(1024, 1)
void silhouette_kernel(const float* __restrict__ tri_tab,
                       float* __restrict__ out) {
    extern __shared__ float smem[];

    const int batch = blockIdx.x >> 7;                 // 128 blocks per batch
    const float* src = tri_tab + (size_t)batch * BATCH_FLOATS;

    // Cooperative LDS fill: 15360 float4 / 1024 threads = 15 each.
    {
        float4* dstv = (float4*)smem;
        const float4* srcv = (const float4*)src;
#pragma unroll
        for (int i = threadIdx.x; i < BATCH_FLOATS / 4; i += 1024)
            dstv[i] = srcv[i];
    }
    __syncthreads();

    const int wave = threadIdx.x >> 5;
    const int lane = threadIdx.x & 31;
    const int group = ((blockIdx.x & 127) << 5) + wave; // 0..4095 per batch
    const int iy  = group >> 4;
    const int ix0 = (group & 15) << 4;

    // B operand: basis (1, qx, qy, qx^2+qy^2) for 16 pixel columns.
    // Lanes 0-15 hold K=0 (v0), K=1 (v1); lanes 16-31 hold K=2, K=3.
    const int pix = lane & 15;
    const float qx = 1.0f - (2.0f * (float)(ix0 + pix) + 1.0f) * (1.0f / (float)IMG);
    const float qy = 1.0f - (2.0f * (float)iy + 1.0f) * (1.0f / (float)IMG);
    const bool hi = lane >= 16;
    v2f bv;
    bv.x = hi ? qy : 1.0f;
    bv.y = hi ? (qx * qx + qy * qy) : qx;

    // A operand addressing: lane L supplies A row M = L&15 (K pair by L>>4).
    const int M = lane & 15;
    const float* aPtr = smem + ((M >> 3) * 32 + (M & 7) * 4 + ((lane >> 4) << 1));
    // Elementwise-phase constants: lanes 0-15 -> even tri, 16-31 -> odd tri.
    const float* cPtr = smem + COEF_FLOATS + ((lane >> 4) << 3);

    float keep = 1.0f;
    const v8f cz = {0.f, 0.f, 0.f, 0.f, 0.f, 0.f, 0.f, 0.f};

    for (int pg = 0; pg < PAIRS / 4; ++pg) {
        // Group-base pointers advance once per group; the k*64 / k*16 indices
        // become 16-bit immediate offsets inside the DS load encodings.
#pragma unroll 2
        for (int k = 0; k < 4; ++k) {
            v2f av = *(const v2f*)(aPtr + k * 64);

            v8f d = __builtin_amdgcn_wmma_f32_16x16x4_f32(
                false, av, false, bv, (short)0, cz, false, false);

            const float* cp = cPtr + k * 16;
            const v4f c0 = *(const v4f*)cp;         // ee0, ee1, ee2, ivn0
            const v2f c1 = *(const v2f*)(cp + 4);   // ivn1, ivn2
            const float ee0 = c0.x, ee1 = c0.y, ee2 = c0.z;
            const float iv0 = c0.w, iv1 = c1.x, iv2 = c1.y;

            const float u0 = d[0], u1 = d[1], u2 = d[2];
            const float x0 = d[3], x1 = d[4], x2 = d[5];
            const float w2a = d[6], w2b = d[7];
            const float w2c = (w2b + u1) + ee1;      // |q-c|^2

            const float t0 = fminf(fmaxf(u0 * iv0, 0.0f), 1.0f);
            const float t1 = fminf(fmaxf(u1 * iv1, 0.0f), 1.0f);
            const float t2 = fminf(fmaxf(u2 * iv2, 0.0f), 1.0f);
            // d2 = |w|^2 - 2 t s + t^2 ee  (u = -2s)
            const float d0 = fmaf(t0, fmaf(t0, ee0, u0), w2a);
            const float d1 = fmaf(t1, fmaf(t1, ee1, u1), w2b);
            const float d2 = fmaf(t2, fmaf(t2, ee2, u2), w2c);
            // max(.,0) commutes with min: clamp cancellation noise once
            const float dmin = fmaxf(fminf(fminf(d0, d1), d2), 0.0f);

            const float mn = fminf(fminf(x0, x1), x2);
            const float mx = fmaxf(fmaxf(x0, x1), x2);
            const bool inside = (mn >= 0.0f) || (mx <= 0.0f);

            // sigmoid(dmin/SIGMA) = rcp(1 + exp2(dmin * EXP_SCALE))
            const float ex = __builtin_amdgcn_exp2f(dmin * EXP_SCALE);
            float om = 1.0f - __builtin_amdgcn_rcpf(1.0f + ex);  // 1 - prob
            om = inside ? om : 1.0f;
            keep *= om;
        }
        aPtr += 256;   // 4 pairs * 64 floats
        cPtr += 64;    // 4 pairs * 16 floats

        // Wave-uniform early exit: keep==0 is absorbing, result unchanged.
        if (!__any(keep != 0.0f)) break;
    }

    // Merge even/odd triangle partial products for the same pixel.
    const float other = __shfl_xor(keep, 16, 32);
    keep *= other;

    if (lane < 16) {
        const float alpha = 1.0f - keep;
        // sigmoid(6a)*2-1 = 2*rcp(1+exp2(-6a*log2e)) - 1
        const float ex = __builtin_amdgcn_exp2f(alpha * OUT_SCALE);
        const float o = fmaf(2.0f, __builtin_amdgcn_rcpf(1.0f + ex), -1.0f);
        out[((size_t)batch * IMG + iy) * IMG + ix0 + lane] = o;
    }
}

// ---------------------------------------------------------------------------
extern "C" void kernel_launch(void* const* d_in, const int* in_sizes, int n_in,
                              void* d_out, int out_size, void* d_ws, size_t ws_size,
                              hipStream_t stream) {
    const float* verts = (const float*)d_in[0];   // (4, 778, 3) f32
    const int*   faces = (const int*)d_in[1];     // (1536, 3) i32
    const int*   flag  = (const int*)d_in[2];     // scalar
    float* out = (float*)d_out;                   // (4, 256, 256) f32
    float* ws  = (float*)d_ws;                    // needs 983 KB

    prep_kernel<<<(NB * NF + 255) / 256, 256, 0, stream>>>(verts, faces, flag, ws);

    const size_t smem_bytes = (size_t)BATCH_FLOATS * sizeof(float);  // 240 KB
    silhouette_kernel<<<512, 1024, smem_bytes, stream>>>(ws, out);
}